// GCNLinkPredictor_75831942578596
// MI455X (gfx1250) — compile-verified
//
#include <hip/hip_runtime.h>
#include <hip/hip_bf16.h>

typedef __attribute__((ext_vector_type(2))) float v2f;
typedef __attribute__((ext_vector_type(8))) float v8f;

#define FEAT 64

// ---------------------------------------------------------------------------
// Hardware f32 atomic add (global_atomic_add_f32), no CAS loop.
// ---------------------------------------------------------------------------
__device__ __forceinline__ void atomAddF(float* p, float v) {
  unsafeAtomicAdd(p, v);
}

// ---------------------------------------------------------------------------
// deg[i] = 1.0 (self loop baseline)
// ---------------------------------------------------------------------------
__global__ void k_init_deg(float* __restrict__ deg, int n) {
  int i = blockIdx.x * blockDim.x + threadIdx.x;
  if (i < n) deg[i] = 1.0f;
}

// deg[col[e]] += 1 for every edge
__global__ void k_count_deg(const int* __restrict__ col, float* __restrict__ deg,
                            int E) {
  int e = blockIdx.x * blockDim.x + threadIdx.x;
  if (e < E) atomAddF(&deg[col[e]], 1.0f);
}

// deg -> deg^-1/2 in place
__global__ void k_rsqrt(float* __restrict__ d, int n) {
  int i = blockIdx.x * blockDim.x + threadIdx.x;
  if (i < n) {
    float v = d[i];
    d[i] = (v > 0.0f) ? rsqrtf(v) : 0.0f;
  }
}

// ---------------------------------------------------------------------------
// H[M,64] = X[M,64] @ W[64,64] via V_WMMA_F32_16X16X4_F32.
// One wave computes one 16x16 tile; 16 K-steps of K=4.
// A 16x4 layout: lanes 0-15 -> M=lane, VGPR0=K0,VGPR1=K1; lanes 16-31 -> K2,K3.
// B 4x16 layout: mirrored (N across lanes, K split across half-waves).
// C/D: VGPR r -> M=r (lanes 0-15) / M=r+8 (lanes 16-31), N=lane&15.
// ---------------------------------------------------------------------------
__global__ void k_gemm_wmma(const float* __restrict__ X,
                            const float* __restrict__ W,
                            float* __restrict__ H, int M) {
  int wave = (int)((blockIdx.x * blockDim.x + threadIdx.x) >> 5);
  int lane = threadIdx.x & 31;
  int mt = wave >> 2;        // 4 N-tiles (64/16)
  int nt = wave & 3;
  if (mt * 16 >= M) return;  // grids are sized exactly; never divergent

  int l15  = lane & 15;
  int kgrp = (lane >> 4) * 2;           // 0 or 2
  const float* xrow = X + (mt * 16 + l15) * FEAT;
  int ncol = nt * 16 + l15;

  v8f acc = {};
#pragma unroll
  for (int kk = 0; kk < 16; ++kk) {
    int k0 = kk * 4 + kgrp;
    v2f a;
    a.x = xrow[k0];
    a.y = xrow[k0 + 1];
    v2f b;
    b.x = W[k0 * FEAT + ncol];
    b.y = W[(k0 + 1) * FEAT + ncol];
    acc = __builtin_amdgcn_wmma_f32_16x16x4_f32(
        /*neg_a=*/false, a, /*neg_b=*/false, b,
        /*c_mod=*/(short)0, acc, /*reuse_a=*/false, /*reuse_b=*/false);
  }

  int om = mt * 16 + (lane >> 4) * 8;
#pragma unroll
  for (int r = 0; r < 8; ++r) {
    H[(om + r) * FEAT + ncol] = acc[r];
  }
}

// ---------------------------------------------------------------------------
// Out[i,:] = b[:] + dinv[i]^2 * H[i,:]   (self-loop term + bias)
// 16 threads per node, float4 each.
// ---------------------------------------------------------------------------
__global__ void k_init_out(const float* __restrict__ H,
                           const float* __restrict__ dinv,
                           const float* __restrict__ bias,
                           float* __restrict__ Out, int n) {
  long long t = (long long)blockIdx.x * blockDim.x + threadIdx.x;
  if (t >= (long long)n * 16) return;
  int node = (int)(t >> 4);
  int q = (int)(t & 15);
  float di = dinv[node];
  float s = di * di;
  float4 h = ((const float4*)(H + (long long)node * FEAT))[q];
  float4 bb = ((const float4*)bias)[q];
  float4 o;
  o.x = bb.x + s * h.x;
  o.y = bb.y + s * h.y;
  o.z = bb.z + s * h.z;
  o.w = bb.w + s * h.w;
  ((float4*)(Out + (long long)node * FEAT))[q] = o;
}

// ---------------------------------------------------------------------------
// Edge scatter: Out[col[e],:] += dinv[row]*dinv[col] * H[row[e],:]
// 16 threads per edge, each handles one float4 (contiguous 256B row).
// Memory-bound: ~0.85 GB of traffic per layer -> ~40-80us at 23.3 TB/s.
// ---------------------------------------------------------------------------
__global__ void k_edge_scatter(const int* __restrict__ row,
                               const int* __restrict__ col,
                               const float* __restrict__ dinv,
                               const float* __restrict__ H,
                               float* __restrict__ Out, int E) {
  long long t = (long long)blockIdx.x * blockDim.x + threadIdx.x;
  long long e = t >> 4;
  if (e >= E) return;
  int q = (int)(t & 15);
  int r = row[e];
  int c = col[e];
  float nrm = dinv[r] * dinv[c];
  float4 v = ((const float4*)(H + (long long)r * FEAT))[q];
  float* op = Out + (long long)c * FEAT + q * 4;
  atomAddF(op + 0, nrm * v.x);
  atomAddF(op + 1, nrm * v.y);
  atomAddF(op + 2, nrm * v.z);
  atomAddF(op + 3, nrm * v.w);
}

// In-place ReLU, float4 vectorized.
__global__ void k_relu(float* __restrict__ A, long long n4) {
  long long t = (long long)blockIdx.x * blockDim.x + threadIdx.x;
  if (t >= n4) return;
  float4 v = ((float4*)A)[t];
  v.x = fmaxf(v.x, 0.0f);
  v.y = fmaxf(v.y, 0.0f);
  v.z = fmaxf(v.z, 0.0f);
  v.w = fmaxf(v.w, 0.0f);
  ((float4*)A)[t] = v;
}

// ---------------------------------------------------------------------------
// Host side
// ---------------------------------------------------------------------------
extern "C" void kernel_launch(void* const* d_in, const int* in_sizes, int n_in,
                              void* d_out, int out_size, void* d_ws, size_t ws_size,
                              hipStream_t stream) {
  const float* x   = (const float*)d_in[0];
  const int*   ei  = (const int*)d_in[1];   // int64 in reference demoted to int32 by JAX defaults
  const float* W1  = (const float*)d_in[2];
  const float* b1  = (const float*)d_in[3];
  const float* W2  = (const float*)d_in[4];
  const float* b2  = (const float*)d_in[5];
  float* out = (float*)d_out;

  const int N = in_sizes[0] / FEAT;       // 100000
  const int E = in_sizes[1] / 2;          // 1600000
  const int* row = ei;                    // edge_index[0]
  const int* col = ei + E;                // edge_index[1]

  // workspace layout (floats): dinv[Npad] | h[N*64] | a1[N*64]
  const int Npad = ((N + 1023) / 1024) * 1024;
  float* dinv = (float*)d_ws;
  float* hbuf = dinv + Npad;
  float* a1   = hbuf + (size_t)N * FEAT;

  const int B = 256;
  const int gN    = (N + B - 1) / B;
  const int gE    = (E + B - 1) / B;
  const int gRow  = (int)(((long long)N * 16 + B - 1) / B);   // 16 thr/node
  const int gEdge = (int)(((long long)E * 16 + B - 1) / B);   // 16 thr/edge
  const int mtiles = (N + 15) / 16;
  const int gGemm = (mtiles * 4 + 7) / 8;                     // 8 waves/block

  // normalization coefficients
  k_init_deg<<<gN, B, 0, stream>>>(dinv, N);
  k_count_deg<<<gE, B, 0, stream>>>(col, dinv, E);
  k_rsqrt<<<gN, B, 0, stream>>>(dinv, N);

  // ---- layer 1: a1 = relu(scatter(norm * (x@W1)[row] -> col) + b1) ----
  k_gemm_wmma<<<gGemm, B, 0, stream>>>(x, W1, hbuf, N);
  k_init_out<<<gRow, B, 0, stream>>>(hbuf, dinv, b1, a1, N);
  k_edge_scatter<<<gEdge, B, 0, stream>>>(row, col, dinv, hbuf, a1, E);
  k_relu<<<gRow, B, 0, stream>>>(a1, (long long)N * 16);

  // ---- layer 2: out = scatter(norm * (a1@W2)[row] -> col) + b2 ----
  k_gemm_wmma<<<gGemm, B, 0, stream>>>(a1, W2, hbuf, N);
  k_init_out<<<gRow, B, 0, stream>>>(hbuf, dinv, b2, out, N);
  k_edge_scatter<<<gEdge, B, 0, stream>>>(row, col, dinv, hbuf, out, E);
}